// GAT_42640435315365
// MI455X (gfx1250) — compile-verified
//
#include <hip/hip_runtime.h>

typedef float v2f __attribute__((ext_vector_type(2)));
typedef float v8f __attribute__((ext_vector_type(8)));

#define NEG_SLOPE 0.2f

// ---------------------------------------------------------------- utilities

__global__ void fill_f32(float* __restrict__ p, long n, float v) {
  long i = (long)blockIdx.x * blockDim.x + threadIdx.x;
  long stride = (long)gridDim.x * blockDim.x;
  for (; i < n; i += stride) p[i] = v;
}

// ------------------------------------------------- fp32 WMMA GEMM (16x16x4)
// C[M,Nd] = A[M,K] @ B[K,Nd], row-major fp32. One wave32 per 16x16 tile.
// Block stages the 16xK A slab in LDS (shared by all waves of the block's
// tile row); B stays in global (tiny, L2-resident).
// A-frag (16x4): lane holds M=lane&15; VGPR j holds K = j + 2*(lane>>4)
// B-frag (4x16): lane holds N=lane&15; VGPR j holds K = j + 2*(lane>>4)
// C/D (16x16):   VGPR r holds M = r + 8*(lane>>4), N = lane&15
__global__ void gemm_wmma_f32_lds(const float* __restrict__ A, const float* __restrict__ B,
                                  float* __restrict__ C, int M, int K, int Nd,
                                  int tnPerBlock) {
  extern __shared__ float As[];                       // 16 * K floats
  int tilesN = Nd >> 4;
  int blocksPerRow = tilesN / tnPerBlock;
  int tm  = blockIdx.x / blocksPerRow;
  int tn0 = (blockIdx.x % blocksPerRow) * tnPerBlock;
  int wib  = threadIdx.x >> 5;
  int lane = threadIdx.x & 31;

  // cooperative b128 fill of the A slab (rows tm*16 .. tm*16+15 contiguous)
  {
    const float4* Av  = (const float4*)(A + (long)tm * 16 * K);
    float4*       Asv = (float4*)As;
    int nvec = (16 * K) >> 2;
    for (int idx = threadIdx.x; idx < nvec; idx += blockDim.x) Asv[idx] = Av[idx];
  }
  __syncthreads();

  int tn   = tn0 + wib;
  int half = lane >> 4;
  int l15  = lane & 15;
  int bcol = tn * 16 + l15;
  v8f acc = {};
  for (int k = 0; k < K; k += 4) {
    int ka = k + 2 * half;
    v2f a, b;
    a.x = As[l15 * K + ka];                           // ds_load
    a.y = As[l15 * K + ka + 1];
    b.x = B[(long)ka * Nd + bcol];
    b.y = B[(long)(ka + 1) * Nd + bcol];
    acc = __builtin_amdgcn_wmma_f32_16x16x4_f32(false, a, false, b,
                                                (short)0, acc, false, false);
  }
  long crow = (long)tm * 16 + half * 8;
#pragma unroll
  for (int r = 0; r < 8; ++r)
    C[(crow + r) * Nd + bcol] = acc[r];
}

// ----------------------------------------- attention logits: wave per (node,head)
__global__ void logits_kernel(const float* __restrict__ hmat, const float* __restrict__ a_src,
                              const float* __restrict__ a_dst, float* __restrict__ als,
                              float* __restrict__ ald, int nnodes, int heads, int ch) {
  int wid  = (int)(((long)blockIdx.x * blockDim.x + threadIdx.x) >> 5);
  int lane = threadIdx.x & 31;
  if (wid >= nnodes * heads) return;
  int node = wid / heads, hh = wid % heads;
  int F = heads * ch;
  float ss = 0.f, dd = 0.f;
  for (int c = lane; c < ch; c += 32) {
    float hv = hmat[(long)node * F + hh * ch + c];
    ss += hv * a_src[hh * ch + c];
    dd += hv * a_dst[hh * ch + c];
  }
#pragma unroll
  for (int off = 16; off > 0; off >>= 1) {
    ss += __shfl_xor(ss, off);
    dd += __shfl_xor(dd, off);
  }
  if (lane == 0) { als[wid] = ss; ald[wid] = dd; }
}

// ------------------------------------------------------------- edge helpers
__device__ __forceinline__ void edge_sd(const int* __restrict__ eidx, int E, int e,
                                        int& s, int& d) {
  if (e < E) { s = eidx[e]; d = eidx[E + e]; }   // edge_index[0]=src, [1]=dst
  else       { s = e - E; d = s; }               // self-loop
}

// pass 1: segment max of leaky-relu logits (thread per edge-head)
__global__ void edge_max_kernel(const int* __restrict__ eidx, int E, int EN,
                                const float* __restrict__ als, const float* __restrict__ ald,
                                float* __restrict__ m, int heads) {
  long t = (long)blockIdx.x * blockDim.x + threadIdx.x;
  if (t >= (long)EN * heads) return;
  int e = (int)(t / heads), hh = (int)(t % heads);
  int s, d; edge_sd(eidx, E, e, s, d);
  float l = als[s * heads + hh] + ald[d * heads + hh];
  l = l > 0.f ? l : NEG_SLOPE * l;
  unsafeAtomicMax(&m[d * heads + hh], l);        // global_atomic_max_num_f32
}

// pass 2 (fused): s[d,h] += ex ; acc[d,:] += h[s,:] * ex
// one wave per edge; lane owns NF = F/32 contiguous channels -> per-lane b128
// gathers, per-lane head index is fixed, no register arrays.
template<int HEADS, int CH>
__global__ void edge_fused_kernel(const int* __restrict__ eidx, int E, int EN,
                                  const float* __restrict__ als, const float* __restrict__ ald,
                                  const float* __restrict__ m, float* __restrict__ ssum,
                                  const float* __restrict__ hsrc, float* __restrict__ accum) {
  constexpr int F  = HEADS * CH;
  constexpr int NF = F / 32;                     // floats per lane (8 or 2)
  int wid  = (int)(((long)blockIdx.x * blockDim.x + threadIdx.x) >> 5);
  int lane = threadIdx.x & 31;
  if (wid >= EN) return;
  int s, d; edge_sd(eidx, E, wid, s, d);
  int j0   = lane * NF;
  int head = j0 / CH;
  float l = als[s * HEADS + head] + ald[d * HEADS + head];
  l = l > 0.f ? l : NEG_SLOPE * l;
  float ex = __expf(l - m[d * HEADS + head]);
  if ((j0 % CH) == 0)                            // one lane per head
    unsafeAtomicAdd(&ssum[d * HEADS + head], ex);
  const float* hp = hsrc  + (long)s * F + j0;
  float*       ap = accum + (long)d * F + j0;
  float v[NF];
#pragma unroll
  for (int k = 0; k < NF; ++k) v[k] = hp[k];     // merges into global_load_b128
#pragma unroll
  for (int k = 0; k < NF; ++k)
    unsafeAtomicAdd(&ap[k], v[k] * ex);          // global_atomic_add_f32
}

// generic fallback (any heads/ch)
__global__ void edge_fused_gen(const int* __restrict__ eidx, int E, int EN,
                               const float* __restrict__ als, const float* __restrict__ ald,
                               const float* __restrict__ m, float* __restrict__ ssum,
                               const float* __restrict__ hsrc, float* __restrict__ accum,
                               int heads, int ch) {
  int wid  = (int)(((long)blockIdx.x * blockDim.x + threadIdx.x) >> 5);
  int lane = threadIdx.x & 31;
  if (wid >= EN) return;
  int s, d; edge_sd(eidx, E, wid, s, d);
  int F = heads * ch;
  for (int j = lane; j < F; j += 32) {
    int head = j / ch;
    float l = als[s * heads + head] + ald[d * heads + head];
    l = l > 0.f ? l : NEG_SLOPE * l;
    float ex = __expf(l - m[d * heads + head]);
    if ((j % ch) == 0) unsafeAtomicAdd(&ssum[d * heads + head], ex);
    unsafeAtomicAdd(&accum[(long)d * F + j], hsrc[(long)s * F + j] * ex);
  }
}

// epilogue: a[n,j] = act( a[n,j] / s[n, j/ch] + bias[j] )
__global__ void norm_bias_act_kernel(float* __restrict__ a, const float* __restrict__ ssum,
                                     const float* __restrict__ bias, long n, int F,
                                     int ch, int heads, int do_elu) {
  long i = (long)blockIdx.x * blockDim.x + threadIdx.x;
  long stride = (long)gridDim.x * blockDim.x;
  for (; i < n; i += stride) {
    int  j    = (int)(i % F);
    long node = i / F;
    float v = a[i] / ssum[node * heads + j / ch] + bias[j];
    if (do_elu) v = v > 0.f ? v : (__expf(v) - 1.f);
    a[i] = v;
  }
}

// ---------------------------------------------------------------- launcher

static inline int cdiv_i(long a, long b) { return (int)((a + b - 1) / b); }

static void launch_gemm(const float* A, const float* B, float* C,
                        int M, int K, int Nd, hipStream_t stream) {
  int tilesM = M / 16, tilesN = Nd / 16;
  int wpb = tilesN >= 8 ? 8 : tilesN;            // waves (= tn tiles) per block
  dim3 grid(tilesM * (tilesN / wpb));
  dim3 block(32 * wpb);
  size_t shmem = (size_t)16 * K * sizeof(float); // A slab
  gemm_wmma_f32_lds<<<grid, block, shmem, stream>>>(A, B, C, M, K, Nd, wpb);
}

extern "C" void kernel_launch(void* const* d_in, const int* in_sizes, int n_in,
                              void* d_out, int out_size, void* d_ws, size_t ws_size,
                              hipStream_t stream) {
  const float* x   = (const float*)d_in[0];
  const int*   eix = (const int*)d_in[1];
  const float* W1  = (const float*)d_in[2];
  const float* as1 = (const float*)d_in[3];
  const float* ad1 = (const float*)d_in[4];
  const float* b1  = (const float*)d_in[5];
  const float* W2  = (const float*)d_in[6];
  const float* as2 = (const float*)d_in[7];
  const float* ad2 = (const float*)d_in[8];
  const float* b2  = (const float*)d_in[9];
  float* out = (float*)d_out;

  const int IN_DIM = 128;
  const int F1     = in_sizes[5];           // heads*hid = 256
  const int HID    = in_sizes[9];           // 64
  const int HEADS  = F1 / HID;              // 4
  const int Nn     = in_sizes[0] / IN_DIM;  // 50000
  const int E      = in_sizes[1] / 2;       // 800000
  const int EN     = E + Nn;                // edges + self loops

  // workspace layout (floats)
  float* acc1 = (float*)d_ws;                    // [Nn, F1] -> h1e after epilogue
  float* h1   = acc1 + (long)Nn * F1;            // [Nn, F1]; reused for h2 [Nn, HID]
  float* sc   = h1 + (long)Nn * F1;
  float* als1 = sc;                              // Nn*HEADS each
  float* ald1 = als1 + (long)Nn * HEADS;
  float* m1   = ald1 + (long)Nn * HEADS;
  float* s1   = m1   + (long)Nn * HEADS;
  float* als2 = s1   + (long)Nn * HEADS;         // Nn each
  float* ald2 = als2 + Nn;
  float* m2   = ald2 + Nn;
  float* s2   = m2   + Nn;
  float* h2   = h1;                              // overlay: h1 dead after layer-1 agg

  const float NEGINF = -__builtin_huge_valf();

  // ---- init accumulators / softmax state
  fill_f32<<<cdiv_i((long)Nn * F1, 256), 256, 0, stream>>>(acc1, (long)Nn * F1, 0.f);
  fill_f32<<<cdiv_i((long)Nn * HEADS, 256), 256, 0, stream>>>(m1, (long)Nn * HEADS, NEGINF);
  fill_f32<<<cdiv_i((long)Nn * HEADS, 256), 256, 0, stream>>>(s1, (long)Nn * HEADS, 0.f);
  fill_f32<<<cdiv_i(Nn, 256), 256, 0, stream>>>(m2, Nn, NEGINF);
  fill_f32<<<cdiv_i(Nn, 256), 256, 0, stream>>>(s2, Nn, 0.f);
  fill_f32<<<cdiv_i((long)Nn * HID, 256), 256, 0, stream>>>(out, (long)Nn * HID, 0.f);

  // ---- layer 1: h1 = x @ W1  (WMMA, LDS-staged A)
  launch_gemm(x, W1, h1, Nn, IN_DIM, F1, stream);
  logits_kernel<<<cdiv_i((long)Nn * HEADS * 32, 256), 256, 0, stream>>>(
      h1, as1, ad1, als1, ald1, Nn, HEADS, HID);
  edge_max_kernel<<<cdiv_i((long)EN * HEADS, 256), 256, 0, stream>>>(
      eix, E, EN, als1, ald1, m1, HEADS);
  if (HEADS == 4 && HID == 64)
    edge_fused_kernel<4, 64><<<cdiv_i((long)EN * 32, 256), 256, 0, stream>>>(
        eix, E, EN, als1, ald1, m1, s1, h1, acc1);
  else
    edge_fused_gen<<<cdiv_i((long)EN * 32, 256), 256, 0, stream>>>(
        eix, E, EN, als1, ald1, m1, s1, h1, acc1, HEADS, HID);
  norm_bias_act_kernel<<<cdiv_i((long)Nn * F1, 256), 256, 0, stream>>>(
      acc1, s1, b1, (long)Nn * F1, F1, HID, HEADS, 1);

  // ---- layer 2: h2 = h1e @ W2  (WMMA, LDS-staged A)
  launch_gemm(acc1, W2, h2, Nn, F1, HID, stream);
  logits_kernel<<<cdiv_i((long)Nn * 32, 256), 256, 0, stream>>>(
      h2, as2, ad2, als2, ald2, Nn, 1, HID);
  edge_max_kernel<<<cdiv_i((long)EN, 256), 256, 0, stream>>>(
      eix, E, EN, als2, ald2, m2, 1);
  if (HID == 64)
    edge_fused_kernel<1, 64><<<cdiv_i((long)EN * 32, 256), 256, 0, stream>>>(
        eix, E, EN, als2, ald2, m2, s2, h2, out);
  else
    edge_fused_gen<<<cdiv_i((long)EN * 32, 256), 256, 0, stream>>>(
        eix, E, EN, als2, ald2, m2, s2, h2, out, 1, HID);
  norm_bias_act_kernel<<<cdiv_i((long)Nn * HID, 256), 256, 0, stream>>>(
      out, s2, b2, (long)Nn * HID, HID, HID, 1, 0);
}